// GraphClassifier_71829033058897
// MI455X (gfx1250) — compile-verified
//
#include <hip/hip_runtime.h>
#include <hip/hip_bf16.h>

typedef __attribute__((ext_vector_type(2))) float v2f;
typedef __attribute__((ext_vector_type(8))) float v8f;

#define N_NODES_C 50000
#define N_EDGES_C 800000
#define N_GRAPHS_C 128
#define FEATS 128
#define N_CLASSES 10

__device__ __forceinline__ void atom_add(float* p, float v) {
    unsafeAtomicAdd(p, v);   // lowers to global_atomic_add_f32
}

// ---------------- degree accumulation ----------------
__global__ __launch_bounds__(256) void degree_kernel(const int* __restrict__ src,
                                                     const int* __restrict__ dst,
                                                     float* __restrict__ dout,
                                                     float* __restrict__ din, int nE) {
    int i = blockIdx.x * blockDim.x + threadIdx.x;
    if (i < nE) {
        atom_add(&dout[src[i]], 1.0f);
        atom_add(&din[dst[i]], 1.0f);
    }
}

// in-place: d = rsqrt(max(d,1))
__global__ __launch_bounds__(256) void rsqrt_kernel(float* __restrict__ d, int n) {
    int i = blockIdx.x * blockDim.x + threadIdx.x;
    if (i < n) d[i] = rsqrtf(fmaxf(d[i], 1.0f));
}

// ---------------- node GEMM via V_WMMA_F32_16X16X4_F32 ----------------
// Y[50000,128] = (diag(scale) * X[50000,128]) @ W[128,128]
// One wave32 per 16x16 output tile; block = 8 waves covering all 8 column tiles.
// f32 WMMA VGPR striping (ISA 7.12.2):
//   A(16x4):  lane m=l%16 holds rows; K = vgpr + 2*(l>=16)
//   B(4x16):  lane n=l%16 holds cols; K = vgpr + 2*(l>=16)
//   C/D(16x16): lane n=l%16; M = vgpr + 8*(l>=16)
__global__ __launch_bounds__(256) void gemm_node_kernel(const float* __restrict__ X,
                                                        const float* __restrict__ scale,
                                                        const float* __restrict__ W,
                                                        float* __restrict__ Y) {
    const int lane  = threadIdx.x & 31;
    const int wave  = threadIdx.x >> 5;      // column tile 0..7
    const int m     = lane & 15;
    const int khalf = lane >> 4;             // 0 or 1
    const int row   = blockIdx.x * 16 + m;   // exact: 50000 = 3125*16
    const int col   = wave * 16 + m;

    const float s = scale[row];
    const float* xrow = X + (size_t)row * FEATS;

    v8f acc = {};
#pragma unroll
    for (int k = 0; k < FEATS; k += 4) {
        const int ka = k + 2 * khalf;
        v2f a, b;
        a.x = xrow[ka] * s;
        a.y = xrow[ka + 1] * s;
        b.x = W[(size_t)ka * FEATS + col];
        b.y = W[(size_t)(ka + 1) * FEATS + col];
        acc = __builtin_amdgcn_wmma_f32_16x16x4_f32(
            /*neg_a=*/false, a, /*neg_b=*/false, b,
            /*c_mod=*/(short)0, acc, /*reuse_a=*/false, /*reuse_b=*/false);
    }

    float* yb = Y + (size_t)(blockIdx.x * 16) * FEATS + wave * 16 + m;
#pragma unroll
    for (int i = 0; i < 8; ++i) {
        yb[(size_t)(i + 8 * khalf) * FEATS] = acc[i];
    }
}

// ---------------- edge gather + scatter-add ----------------
// wave per edge; lane moves 4 contiguous floats (float4 coalesced gather,
// 4x global_atomic_add_f32 scatter).
__global__ __launch_bounds__(256) void scatter_kernel(const float* __restrict__ Xp,
                                                      const int* __restrict__ src,
                                                      const int* __restrict__ dst,
                                                      float* __restrict__ agg, int nE) {
    int gid  = blockIdx.x * blockDim.x + threadIdx.x;
    int e    = gid >> 5;
    int lane = gid & 31;
    if (e < nE) {
        const int s = src[e];
        const int d = dst[e];
        const float4 v = *(const float4*)(Xp + (size_t)s * FEATS + lane * 4);
        float* a = agg + (size_t)d * FEATS + lane * 4;
        atom_add(a + 0, v.x);
        atom_add(a + 1, v.y);
        atom_add(a + 2, v.z);
        atom_add(a + 3, v.w);
    }
}

// ---------------- epilogue: h = relu(agg * in_isqrt + b) ----------------
__global__ __launch_bounds__(256) void finish_kernel(const float* __restrict__ agg,
                                                     const float* __restrict__ isq_in,
                                                     const float* __restrict__ bias,
                                                     float* __restrict__ hout, int total) {
    int gid = blockIdx.x * blockDim.x + threadIdx.x;
    if (gid < total) {
        int node = gid >> 7;
        int f    = gid & 127;
        float v  = agg[gid] * isq_in[node] + bias[f];
        hout[gid] = fmaxf(v, 0.0f);
    }
}

// ---------------- readout ----------------
__global__ __launch_bounds__(256) void count_kernel(const int* __restrict__ gids,
                                                    float* __restrict__ counts, int n) {
    int i = blockIdx.x * blockDim.x + threadIdx.x;
    if (i < n) atom_add(&counts[gids[i]], 1.0f);
}

__global__ __launch_bounds__(256) void pool_kernel(const float* __restrict__ h,
                                                   const int* __restrict__ gids,
                                                   float* __restrict__ hg, int nNodes) {
    int gid  = blockIdx.x * blockDim.x + threadIdx.x;
    int node = gid >> 5;
    int lane = gid & 31;
    if (node < nNodes) {
        const int g = gids[node];
        const float4 v = *(const float4*)(h + (size_t)node * FEATS + lane * 4);
        float* a = hg + (size_t)g * FEATS + lane * 4;
        atom_add(a + 0, v.x);
        atom_add(a + 1, v.y);
        atom_add(a + 2, v.z);
        atom_add(a + 3, v.w);
    }
}

__global__ __launch_bounds__(256) void head_kernel(const float* __restrict__ hg,
                                                   const float* __restrict__ counts,
                                                   const float* __restrict__ Wr,
                                                   float* __restrict__ out) {
    int gid = blockIdx.x * blockDim.x + threadIdx.x;
    if (gid < N_GRAPHS_C * N_CLASSES) {
        int g = gid / N_CLASSES;
        int c = gid % N_CLASSES;
        float inv = 1.0f / fmaxf(counts[g], 1.0f);
        float acc = 0.0f;
        for (int f = 0; f < FEATS; ++f)
            acc += hg[(size_t)g * FEATS + f] * Wr[(size_t)f * N_CLASSES + c];
        out[gid] = acc * inv;
    }
}

extern "C" void kernel_launch(void* const* d_in, const int* in_sizes, int n_in,
                              void* d_out, int out_size, void* d_ws, size_t ws_size,
                              hipStream_t stream) {
    const float* feat = (const float*)d_in[0];
    const int*   src  = (const int*)d_in[1];
    const int*   dst  = (const int*)d_in[2];
    const int*   gids = (const int*)d_in[3];
    const float* W0   = (const float*)d_in[4];
    const float* b0   = (const float*)d_in[5];
    const float* W1   = (const float*)d_in[6];
    const float* b1   = (const float*)d_in[7];
    const float* W2   = (const float*)d_in[8];
    const float* b2   = (const float*)d_in[9];
    const float* Wr   = (const float*)d_in[10];
    float* out = (float*)d_out;

    const int nN = in_sizes[0] / FEATS;   // 50000
    const int nE = in_sizes[1];           // 800000
    const size_t nf = (size_t)nN * FEATS; // 6.4M

    // workspace layout (floats)
    float* ws      = (float*)d_ws;
    float* isq_out = ws;                        // nN (deg -> rsqrt in place)
    float* isq_in  = isq_out + nN;              // nN
    float* counts  = isq_in + nN;               // 128
    float* hg      = counts + 128;              // 128*128
    float* bufA    = hg + N_GRAPHS_C * FEATS;   // nf : Xp
    float* bufB    = bufA + nf;                 // nf : h
    float* agg     = bufB + nf;                 // nf

    // ---- degrees / norms ----
    hipMemsetAsync(isq_out, 0, 2 * (size_t)nN * sizeof(float), stream);
    degree_kernel<<<(nE + 255) / 256, 256, 0, stream>>>(src, dst, isq_out, isq_in, nE);
    rsqrt_kernel<<<(nN + 255) / 256, 256, 0, stream>>>(isq_out, nN);
    rsqrt_kernel<<<(nN + 255) / 256, 256, 0, stream>>>(isq_in, nN);

    const int gemm_blocks    = nN / 16;                 // 3125, exact
    const int scatter_blocks = (nE * 32 + 255) / 256;   // 100000
    const int finish_blocks  = ((int)nf + 255) / 256;   // 25000

    const float* hin = feat;
    const float* biases[3] = { b0, b1, b2 };
    const float* weights[3] = { W0, W1, W2 };
    for (int layer = 0; layer < 3; ++layer) {
        // Xp = (diag(out_isqrt) * hin) @ W
        gemm_node_kernel<<<gemm_blocks, 256, 0, stream>>>(hin, isq_out, weights[layer], bufA);
        // agg = scatter_add over edges
        hipMemsetAsync(agg, 0, nf * sizeof(float), stream);
        scatter_kernel<<<scatter_blocks, 256, 0, stream>>>(bufA, src, dst, agg, nE);
        // h = relu(agg * in_isqrt + b)
        finish_kernel<<<finish_blocks, 256, 0, stream>>>(agg, isq_in, biases[layer], bufB, (int)nf);
        hin = bufB;
    }

    // ---- readout ----
    hipMemsetAsync(counts, 0, (128 + N_GRAPHS_C * FEATS) * sizeof(float), stream);
    count_kernel<<<(nN + 255) / 256, 256, 0, stream>>>(gids, counts, nN);
    pool_kernel<<<(nN * 32 + 255) / 256, 256, 0, stream>>>(bufB, gids, hg, nN);
    head_kernel<<<(N_GRAPHS_C * N_CLASSES + 255) / 256, 256, 0, stream>>>(hg, counts, Wr, out);
}